// AdaptiveAntiAlias_25881472926182
// MI455X (gfx1250) — compile-verified
//
#include <hip/hip_runtime.h>

// AdaptiveAntiAlias: fused 5x5 Gaussian (zero-pad conv) + 5x5 bilateral
// (reflect pad), out = 0.6*gauss + 0.4*bilateral.  Images: (8,3,512,512) f32.
//
// Compute-bound (~24 v_exp_f32 + ~150 VALU per pixel); strategy:
//  * stage reflect-padded 36x36 tile in LDS via CDNA5 async global->LDS
//    copies (GLOBAL_LOAD_ASYNC_TO_LDS_B32, ASYNCcnt) + s_wait_asynccnt
//  * hardware v_exp_f32; spatial weight folded into the exponent:
//    SW*exp2(d^2*C) = exp2(fma(d*C, d, log2(SW))) -- log2(SW) is constant
//  * center tap (d=0, SW=1) contributes exactly iw=1 -> hoisted out
//  * register 5x5 sliding window: 4 output rows per thread (~40 LDS reads
//    per 4 outputs instead of 100)
//  * Gaussian weights separable: GW*rowmask*colmask = a[dy]*b[dx] ->
//    row sums (25 fma + 5 fma) instead of 75 VALU
//  * branch-free row/col masks emulate the conv's zero padding while the
//    LDS tile holds the bilateral's reflect padding

#define IMG_H 512
#define IMG_W 512
#define TILE  32
#define LDSW  36            // TILE + 2*PAD
#define NTHR  256
#define EXP2C (-72.134752044448169f)   // -(1/(2*0.1^2)) * log2(e)

typedef __attribute__((address_space(1))) int gint;   // global (as1)
typedef __attribute__((address_space(3))) int lint;   // LDS (as3)

__device__ __forceinline__ void g2l_async(const float* g, float* l) {
#if __has_builtin(__builtin_amdgcn_global_load_async_to_lds_b32)
  __builtin_amdgcn_global_load_async_to_lds_b32(
      (gint*)(const_cast<float*>(g)), (lint*)l, /*offset=*/0, /*cpol=*/0);
#else
  *l = *g;
#endif
}

__device__ __forceinline__ void wait_async_zero() {
#if __has_builtin(__builtin_amdgcn_s_wait_asynccnt)
  __builtin_amdgcn_s_wait_asynccnt(0);
#endif
}

__global__ __launch_bounds__(NTHR) void
AdaptiveAntiAlias_kernel(const float* __restrict__ in, float* __restrict__ out)
{
  __shared__ float tile[LDSW * LDSW];

  const int plane = blockIdx.z;                 // batch*channel plane
  const int tx0   = blockIdx.x * TILE;          // tile origin (output coords)
  const int ty0   = blockIdx.y * TILE;
  const float* src = in  + (size_t)plane * (IMG_H * IMG_W);
  float*       dst = out + (size_t)plane * (IMG_H * IMG_W);
  const int tid = threadIdx.x;

  // ---- stage reflect-padded 36x36 tile into LDS (async per-lane copies) ----
  for (int i = tid; i < LDSW * LDSW; i += NTHR) {
    int lr = i / LDSW;
    int lc = i - lr * LDSW;
    int gy = ty0 + lr - 2;
    int gx = tx0 + lc - 2;
    gy = gy < 0 ? -gy : gy;  gy = gy >= IMG_H ? 2 * IMG_H - 2 - gy : gy;
    gx = gx < 0 ? -gx : gx;  gx = gx >= IMG_W ? 2 * IMG_W - 2 - gx : gx;
    g2l_async(src + gy * IMG_W + gx, &tile[i]);
  }
  wait_async_zero();
  __syncthreads();

  // ---- constant tables (fully constant-folded) ----
  // 1D spatial kernel and its log2 (SW = exp(-r^2/2); log2(SW) = -r^2/2*log2e)
  const float w1d[5] = {0.13533528323661270f, 0.60653065971263342f, 1.0f,
                        0.60653065971263342f, 0.13533528323661270f};
  const float l1d[5] = {-2.8853900817779268f, -0.72134752044448169f, 0.0f,
                        -0.72134752044448169f, -2.8853900817779268f};
  float ssum = 0.0f;
#pragma unroll
  for (int i = 0; i < 5; ++i)
#pragma unroll
    for (int j = 0; j < 5; ++j) ssum += w1d[i] * w1d[j];
  const float ginv = 1.0f / ssum;   // Gaussian normalization (conv kernel g/g.sum)

  // ---- per-thread geometry: 1 column, 4 consecutive output rows ----
  const int x  = tid & (TILE - 1);        // 0..31
  const int r0 = (tid >> 5) << 2;         // 0,4,...,28

  // separable Gaussian column factors incl. zero-pad column mask (per thread)
  float bx[5];
#pragma unroll
  for (int dx = 0; dx < 5; ++dx) {
    int gx = tx0 + x + dx - 2;
    bx[dx] = (gx >= 0 && gx < IMG_W) ? w1d[dx] : 0.0f;
  }

  // 5x5 register window over LDS rows r0..r0+4
  float win[5][5];
#pragma unroll
  for (int dy = 0; dy < 5; ++dy)
#pragma unroll
    for (int dx = 0; dx < 5; ++dx)
      win[dy][dx] = tile[(r0 + dy) * LDSW + x + dx];

#pragma unroll
  for (int k = 0; k < 4; ++k) {
    const int pr = r0 + k;                // output row within tile

    // separable Gaussian row factors incl. zero-pad row mask (per pixel)
    float ay[5];
#pragma unroll
    for (int dy = 0; dy < 5; ++dy) {
      int gy = ty0 + pr + dy - 2;
      ay[dy] = (gy >= 0 && gy < IMG_H) ? w1d[dy] * ginv : 0.0f;
    }

    const float c = win[2][2];            // center pixel (in-range => exact)
    // center tap: d=0, SW=1 -> weight exactly 1; hoist it out of the loop
    float gsum = 0.0f, acc = c, wsum = 1.0f;
#pragma unroll
    for (int dy = 0; dy < 5; ++dy) {
      float rs = 0.0f;                    // Gaussian row sum
#pragma unroll
      for (int dx = 0; dx < 5; ++dx) {
        const float p = win[dy][dx];
        rs = fmaf(bx[dx], p, rs);
        if (!(dy == 2 && dx == 2)) {
          const float d  = p - c;
          // SW * exp2(d*d*C) == exp2(fma(d*C, d, log2(SWy)+log2(SWx)))
          const float iw = __builtin_amdgcn_exp2f(
              fmaf(d * EXP2C, d, l1d[dy] + l1d[dx]));   // v_exp_f32
          wsum += iw;
          acc   = fmaf(iw, p, acc);
        }
      }
      gsum = fmaf(ay[dy], rs, gsum);
    }

    const float bil = acc * __builtin_amdgcn_rcpf(wsum + 1e-8f);
    dst[(ty0 + pr) * IMG_W + tx0 + x] = fmaf(0.6f, gsum, 0.4f * bil);

    if (k < 3) {                          // slide window down one row
#pragma unroll
      for (int dy = 0; dy < 4; ++dy)
#pragma unroll
        for (int dx = 0; dx < 5; ++dx)
          win[dy][dx] = win[dy + 1][dx];
#pragma unroll
      for (int dx = 0; dx < 5; ++dx)
        win[4][dx] = tile[(pr + 5) * LDSW + x + dx];
    }
  }
}

extern "C" void kernel_launch(void* const* d_in, const int* in_sizes, int n_in,
                              void* d_out, int out_size, void* d_ws, size_t ws_size,
                              hipStream_t stream) {
  (void)n_in; (void)out_size; (void)d_ws; (void)ws_size;
  const float* images = (const float*)d_in[0];
  float* out = (float*)d_out;
  const int planes = in_sizes[0] / (IMG_H * IMG_W);   // B*C = 24
  dim3 grid(IMG_W / TILE, IMG_H / TILE, planes);
  AdaptiveAntiAlias_kernel<<<grid, NTHR, 0, stream>>>(images, out);
}